// GPT2Attention_76888504533121
// MI455X (gfx1250) — compile-verified
//
#include <hip/hip_runtime.h>
#include <math.h>

// ---------------------------------------------------------------------------
// GPT-2 attention forward for MI455X (gfx1250), fp32 via V_WMMA_F32_16X16X4_F32
// + Tensor Data Mover (tensor_load_to_lds) double-buffered GEMM staging.
// Outputs (flat, concatenated): attn_output[B,S,D], attn_weights[B,H,S,S],
//                               k[B,H,S,hd], v[B,H,S,hd]
// ---------------------------------------------------------------------------

typedef float v2f __attribute__((ext_vector_type(2)));
typedef float v8f __attribute__((ext_vector_type(8)));
typedef unsigned int v4u __attribute__((ext_vector_type(4)));
typedef int v4i __attribute__((ext_vector_type(4)));
typedef int v8i __attribute__((ext_vector_type(8)));

#define BB 2
#define SS 2048
#define DD 1024
#define HH 16
#define HD 64
#define NROWS (BB * SS)  // 4096

#if defined(__HIP_DEVICE_COMPILE__) && \
    __has_builtin(__builtin_amdgcn_tensor_load_to_lds)
#define HAVE_TDM 1
#else
#define HAVE_TDM 0
#endif

// D = A(16x4) * B(4x16) + C, fp32.
static __device__ __forceinline__ v8f wmma_f32(v2f a, v2f b, v8f c) {
  return __builtin_amdgcn_wmma_f32_16x16x4_f32(false, a, false, b, (short)0, c,
                                               false, false);
}

#if HAVE_TDM
// Raw LDS byte address = low 32 bits of the flat address (LDS aperture keeps
// the offset in addr[31:0] per the ISA aperture mapping).
static __device__ __forceinline__ unsigned lds_addr_of(const void* p) {
  return (unsigned)(uintptr_t)p;
}

// 2D tile load via TDM.  tile_w elements per row (x4B, multiple of 4 DWORDs),
// tile_h rows, global row stride in elements.  LDS destination gets
// pad_amount DWORDs of padding after every 2^(pad_interval+1) DWORDs.
static __device__ __forceinline__ void tdm_load_2d(const float* gsrc,
                                                   unsigned lds_byte,
                                                   int tile_w, int tile_h,
                                                   int stride_elems,
                                                   int pad_interval_code,
                                                   int pad_amount_code) {
  const unsigned long long ga = (unsigned long long)(uintptr_t)gsrc;
  v4u g0;
  g0.x = 1u;  // count=1, is_restore=0, gather off
  g0.y = lds_byte;
  g0.z = (unsigned)(ga & 0xFFFFFFFFu);                        // addr[31:0]
  g0.w = (unsigned)((ga >> 32) & 0x01FFFFFFu) | (2u << 30);   // addr[56:32]|type=2

  v8i g1;
  g1[0] = (2 << 16)                       // data_size = 4 bytes
          | (1 << 20)                     // pad_enable
          | (pad_interval_code << 22)     // 2^(c+1) DWORDs per row
          | (pad_amount_code << 25);      // (c+1) DWORDs pad
  g1[1] = (stride_elems & 0xFFFF) << 16;  // tensor_dim0[15:0] (row length)
  g1[2] = ((stride_elems >> 16) & 0xFFFF) |
          ((1 << 20 & 0xFFFF) << 16);     // tensor_dim1[15:0]
  g1[3] = (((1 << 20) >> 16) & 0xFFFF) |  // tensor_dim1[31:16]
          ((tile_w & 0xFFFF) << 16);      // tile_dim0
  g1[4] = (tile_h & 0xFFFF);              // tile_dim1 (tile_dim2 = 0 -> 2D)
  g1[5] = stride_elems;                   // tensor_dim0_stride[31:0]
  g1[6] = 0;                              // stride0[47:32] | stride1[15:0]
  g1[7] = 0;                              // stride1[47:16]

  const v4i z4 = {0, 0, 0, 0};
#if __clang_major__ >= 23
  const v8i z8 = {0, 0, 0, 0, 0, 0, 0, 0};
  __builtin_amdgcn_tensor_load_to_lds(g0, g1, z4, z4, z8, 0);
#else
  __builtin_amdgcn_tensor_load_to_lds(g0, g1, z4, z4, 0);
#endif
}
#endif  // HAVE_TDM

// ---------------------------------------------------------------------------
// GEMM: Y = X[NROWS,DD] @ W[DD,DD] + bias.  blockIdx.z selects one of up to
// three weight/dst sets (fused QKV).  HEADSPLIT writes [B,H,S,hd] layout.
// Block tile 64x64, K-step 16, 8 waves; each wave owns a 16x32 tile.
// TDM double-buffers the LDS tiles (wave 0 drives the DMA; TENSORcnt is
// per-wave, so publish with __syncthreads()).
// ---------------------------------------------------------------------------
template <bool HEADSPLIT>
__global__ __launch_bounds__(256) void gemm_proj(
    const float* __restrict__ X, const float* __restrict__ W0,
    const float* __restrict__ W1, const float* __restrict__ W2,
    const float* __restrict__ b0, const float* __restrict__ b1,
    const float* __restrict__ b2, float* __restrict__ d0,
    float* __restrict__ d1, float* __restrict__ d2) {
  const float* W = (blockIdx.z == 0) ? W0 : ((blockIdx.z == 1) ? W1 : W2);
  const float* bias = (blockIdx.z == 0) ? b0 : ((blockIdx.z == 1) ? b1 : b2);
  float* dst = (blockIdx.z == 0) ? d0 : ((blockIdx.z == 1) ? d1 : d2);

  // Row strides 20/68 DWORDs: avoid bank conflicts, match TDM pad (4 DW).
  __shared__ float As[2][64][20];  // [buf][row][K 16 + 4 pad]
  __shared__ float Bs[2][16][68];  // [buf][K][col 64 + 4 pad]

  const int tid = threadIdx.x;
  const int wave = tid >> 5;
  const int lane = tid & 31;
  const int l16 = lane & 15;
  const int half = lane >> 4;

  const int mBase = blockIdx.y * 64;
  const int nBase = blockIdx.x * 64;
  const int wm = (wave & 3) * 16;   // wave M offset in block
  const int wn = (wave >> 2) * 32;  // wave N offset in block

  v8f acc0 = {};
  v8f acc1 = {};

#if HAVE_TDM
  // Preload tile 0 into buffer 0.
  if (wave == 0) {
    tdm_load_2d(X + (size_t)mBase * DD, lds_addr_of(&As[0][0][0]),
                /*w=*/16, /*h=*/64, DD, /*interval 16DW*/ 3, /*pad 4DW*/ 3);
    tdm_load_2d(W + nBase, lds_addr_of(&Bs[0][0][0]),
                /*w=*/64, /*h=*/16, DD, /*interval 64DW*/ 5, /*pad 4DW*/ 3);
    __builtin_amdgcn_s_wait_tensorcnt(0);
  }
  __syncthreads();

  int cur = 0;
  for (int k0 = 0; k0 < DD; k0 += 16) {
    if (wave == 0 && (k0 + 16) < DD) {  // kick DMA for next tile
      tdm_load_2d(X + (size_t)mBase * DD + (k0 + 16),
                  lds_addr_of(&As[cur ^ 1][0][0]), 16, 64, DD, 3, 3);
      tdm_load_2d(W + (size_t)(k0 + 16) * DD + nBase,
                  lds_addr_of(&Bs[cur ^ 1][0][0]), 64, 16, DD, 5, 3);
    }
#pragma unroll
    for (int ks = 0; ks < 4; ++ks) {
      v2f a;  // A frag: row = l16, K = ks*4 + {2*half, 2*half+1}
      a.x = As[cur][wm + l16][ks * 4 + 2 * half + 0];
      a.y = As[cur][wm + l16][ks * 4 + 2 * half + 1];
      v2f bA, bB;  // B frag: col = l16, same K half-split
      bA.x = Bs[cur][ks * 4 + 2 * half + 0][wn + l16];
      bA.y = Bs[cur][ks * 4 + 2 * half + 1][wn + l16];
      bB.x = Bs[cur][ks * 4 + 2 * half + 0][wn + 16 + l16];
      bB.y = Bs[cur][ks * 4 + 2 * half + 1][wn + 16 + l16];
      acc0 = wmma_f32(a, bA, acc0);
      acc1 = wmma_f32(a, bB, acc1);
    }
    __syncthreads();  // everyone done reading buffer `cur`
    if (wave == 0) __builtin_amdgcn_s_wait_tensorcnt(0);
    __syncthreads();  // next buffer visible to all waves
    cur ^= 1;
  }
#else
  // Fallback: cooperative float4 global->LDS staging.
  const int arow = tid >> 2;        // 0..63
  const int acol = (tid & 3) * 4;   // 0,4,8,12
  const int brow = tid >> 4;        // 0..15
  const int bcol = (tid & 15) * 4;  // 0..60
  for (int k0 = 0; k0 < DD; k0 += 16) {
    if (k0 + 16 < DD) {
      __builtin_prefetch(X + (size_t)(mBase + arow) * DD + (k0 + 16) + acol, 0, 1);
      __builtin_prefetch(W + (size_t)(k0 + 16 + brow) * DD + nBase + bcol, 0, 1);
    }
    const float4 av = *(const float4*)(X + (size_t)(mBase + arow) * DD + k0 + acol);
    const float4 bv = *(const float4*)(W + (size_t)(k0 + brow) * DD + nBase + bcol);
    __syncthreads();
    *(float4*)(&As[0][arow][acol]) = av;
    *(float4*)(&Bs[0][brow][bcol]) = bv;
    __syncthreads();
#pragma unroll
    for (int ks = 0; ks < 4; ++ks) {
      v2f a;
      a.x = As[0][wm + l16][ks * 4 + 2 * half + 0];
      a.y = As[0][wm + l16][ks * 4 + 2 * half + 1];
      v2f bA, bB;
      bA.x = Bs[0][ks * 4 + 2 * half + 0][wn + l16];
      bA.y = Bs[0][ks * 4 + 2 * half + 1][wn + l16];
      bB.x = Bs[0][ks * 4 + 2 * half + 0][wn + 16 + l16];
      bB.y = Bs[0][ks * 4 + 2 * half + 1][wn + 16 + l16];
      acc0 = wmma_f32(a, bA, acc0);
      acc1 = wmma_f32(a, bB, acc1);
    }
  }
#endif

  // Epilogue: C/D layout row = r + 8*half, col = l16
  const int rowTop = mBase + wm;
#pragma unroll
  for (int nt = 0; nt < 2; ++nt) {
    v8f acc = nt ? acc1 : acc0;
    const int col = nBase + wn + nt * 16 + l16;
    const float bval = bias[col];
#pragma unroll
    for (int r = 0; r < 8; ++r) {
      const int row = rowTop + r + 8 * half;
      const float val = acc[r] + bval;
      if (HEADSPLIT) {
        const int b = row >> 11;      // / S
        const int s = row & (SS - 1);
        const int h = col >> 6;       // / hd
        const int c = col & (HD - 1);
        dst[(((size_t)(b * HH + h)) * SS + s) * HD + c] = val;
      } else {
        dst[(size_t)row * DD + col] = val;
      }
    }
  }
}

// ---------------------------------------------------------------------------
// Pass 1: raw scores S = (Q/8) K^T with causal -10000 mask, streamed to the
// attn_weights region, plus online softmax stats (row max m, row sum l).
// One wave owns 16 query rows; 8 waves/block -> 128 rows per block.
// ---------------------------------------------------------------------------
__global__ __launch_bounds__(256) void attn_scores(
    const float* __restrict__ Q,  // [B*H, S, hd] (workspace)
    const float* __restrict__ K,  // [B*H, S, hd] (output region)
    float* __restrict__ Wraw,     // [B*H, S, S] raw scores (output region)
    float* __restrict__ Mrow, float* __restrict__ Lrow) {
  const int tid = threadIdx.x;
  const int wave = tid >> 5;
  const int lane = tid & 31;
  const int l16 = lane & 15;
  const int half = lane >> 4;

  const int bh = blockIdx.y;
  const int qbase = blockIdx.x * 128 + wave * 16;
  const float* Qh = Q + (size_t)bh * SS * HD;
  const float* Kh = K + (size_t)bh * SS * HD;
  float* Wh = Wraw + (size_t)bh * SS * SS;

  // Q tile 16x64 in A-fragment form, 1/sqrt(hd)=1/8 folded in
  v2f qa[16];
#pragma unroll
  for (int ks = 0; ks < 16; ++ks) {
    const float* p = Qh + (size_t)(qbase + l16) * HD + ks * 4 + 2 * half;
    qa[ks].x = p[0] * 0.125f;
    qa[ks].y = p[1] * 0.125f;
  }

  float rm[8], rl[8];
#pragma unroll
  for (int v = 0; v < 8; ++v) {
    rm[v] = -3.0e38f;
    rl[v] = 0.0f;
  }

  const int qhi = qbase + 15;
  const int jmax = qhi >> 4;  // causal: only key blocks <= diagonal
  for (int j = 0; j <= jmax; ++j) {
    const int kb = j * 16;
    v8f s = {};
#pragma unroll
    for (int ks = 0; ks < 16; ++ks) {
      // B = K^T: B[d][key] = K[key][d]; col(key) = kb + l16
      const float* p = Kh + (size_t)(kb + l16) * HD + ks * 4 + 2 * half;
      v2f b;
      b.x = p[0];
      b.y = p[1];
      s = wmma_f32(qa[ks], b, s);
    }
#pragma unroll
    for (int v = 0; v < 8; ++v) {
      const int gq = qbase + v + 8 * half;
      const int gk = kb + l16;
      float x = s[v];
      x = (gk > gq) ? -10000.0f : x;  // causal mask, reference semantics
      Wh[(size_t)gq * SS + gk] = x;   // stream raw score
      // 16-lane butterfly reductions (each half = one set of 8 rows)
      float xm = x;
      xm = fmaxf(xm, __shfl_xor(xm, 1, 32));
      xm = fmaxf(xm, __shfl_xor(xm, 2, 32));
      xm = fmaxf(xm, __shfl_xor(xm, 4, 32));
      xm = fmaxf(xm, __shfl_xor(xm, 8, 32));
      const float newm = fmaxf(rm[v], xm);
      float pv = __expf(x - newm);  // exp(-1e4 - m) underflows to exactly 0
      float ps = pv;
      ps += __shfl_xor(ps, 1, 32);
      ps += __shfl_xor(ps, 2, 32);
      ps += __shfl_xor(ps, 4, 32);
      ps += __shfl_xor(ps, 8, 32);
      rl[v] = rl[v] * __expf(rm[v] - newm) + ps;
      rm[v] = newm;
    }
  }

  if (l16 == 0) {
#pragma unroll
    for (int v = 0; v < 8; ++v) {
      const int gq = qbase + v + 8 * half;
      Mrow[(size_t)bh * SS + gq] = rm[v];
      Lrow[(size_t)bh * SS + gq] = rl[v];
    }
  }
}

// ---------------------------------------------------------------------------
// Pass 2: re-read raw scores once, normalize in place (final attn_weights),
// and accumulate O = P @ V with WMMA.  Fully-masked tiles are zero-filled.
// ---------------------------------------------------------------------------
__global__ __launch_bounds__(256) void attn_pv(
    const float* __restrict__ V,  // [B*H, S, hd] (output region)
    float* __restrict__ Wbuf,     // raw in -> normalized weights out (in place)
    const float* __restrict__ Mrow, const float* __restrict__ Lrow,
    float* __restrict__ O) {      // [B, S, D] workspace (c_proj input)
  const int tid = threadIdx.x;
  const int wave = tid >> 5;
  const int lane = tid & 31;
  const int l16 = lane & 15;
  const int half = lane >> 4;

  const int bh = blockIdx.y;
  const int b = bh / HH;
  const int h = bh % HH;
  const int qbase = blockIdx.x * 128 + wave * 16;

  float* Wh = Wbuf + (size_t)bh * SS * SS;
  const float* Vh = V + (size_t)bh * SS * HD;

  const float m_r = Mrow[(size_t)bh * SS + qbase + l16];
  const float inv_l = 1.0f / Lrow[(size_t)bh * SS + qbase + l16];

  v8f acc[4] = {{}, {}, {}, {}};
  const int qhi = qbase + 15;

  for (int j = 0; j < SS / 16; ++j) {
    const int kb = j * 16;
    float* rowp = Wh + (size_t)(qbase + l16) * SS + kb + 2 * half;
    if (kb <= qhi) {  // wave-uniform branch: EXEC stays all-1s for WMMA
      v2f pa[4];
#pragma unroll
      for (int ks = 0; ks < 4; ++ks) {
        float2 rv = *(const float2*)(rowp + ks * 4);
        float2 pv2;
        pv2.x = __expf(rv.x - m_r) * inv_l;  // masked (-1e4) -> exactly 0
        pv2.y = __expf(rv.y - m_r) * inv_l;
        *(float2*)(rowp + ks * 4) = pv2;     // final normalized weight
        pa[ks].x = pv2.x;                    // doubles as A-fragment of P
        pa[ks].y = pv2.y;
      }
#pragma unroll
      for (int ks = 0; ks < 4; ++ks) {
        const float* vp0 = Vh + (size_t)(kb + ks * 4 + 2 * half + 0) * HD;
        const float* vp1 = Vh + (size_t)(kb + ks * 4 + 2 * half + 1) * HD;
#pragma unroll
        for (int nt = 0; nt < 4; ++nt) {
          v2f bf;  // B frag: rows = key k, cols = head-dim
          bf.x = vp0[nt * 16 + l16];
          bf.y = vp1[nt * 16 + l16];
          acc[nt] = wmma_f32(pa[ks], bf, acc[nt]);
        }
      }
    } else {
      const float2 z = {0.0f, 0.0f};  // softmax of fully-masked cols == 0
#pragma unroll
      for (int ks = 0; ks < 4; ++ks) *(float2*)(rowp + ks * 4) = z;
    }
  }

  // O tile 16x64 -> [B,S,D] merged-head layout
#pragma unroll
  for (int nt = 0; nt < 4; ++nt) {
#pragma unroll
    for (int v = 0; v < 8; ++v) {
      const int gq = qbase + v + 8 * half;
      O[((size_t)(b * SS + gq)) * DD + h * HD + nt * 16 + l16] = acc[nt][v];
    }
  }
}

// ---------------------------------------------------------------------------
// Host-side launcher
// ---------------------------------------------------------------------------
extern "C" void kernel_launch(void* const* d_in, const int* in_sizes, int n_in,
                              void* d_out, int out_size, void* d_ws,
                              size_t ws_size, hipStream_t stream) {
  const float* hs = (const float*)d_in[0];
  const float* wq = (const float*)d_in[1];
  const float* bq = (const float*)d_in[2];
  const float* wk = (const float*)d_in[3];
  const float* bk = (const float*)d_in[4];
  const float* wv = (const float*)d_in[5];
  const float* bv = (const float*)d_in[6];
  const float* wc = (const float*)d_in[7];
  const float* bc = (const float*)d_in[8];

  const size_t N_ATTN = (size_t)BB * SS * DD;    // 4,194,304
  const size_t N_W = (size_t)BB * HH * SS * SS;  // 134,217,728

  float* out = (float*)d_out;
  float* attn_out = out;         // [B,S,D]
  float* attn_w = out + N_ATTN;  // [B,H,S,S]
  float* k_out = attn_w + N_W;   // [B,H,S,hd]
  float* v_out = k_out + N_ATTN; // [B,H,S,hd]

  float* ws = (float*)d_ws;
  float* Qws = ws;                // [B,H,S,hd]
  float* Ows = ws + N_ATTN;       // [B,S,D] pre-projection output
  float* Mrow = ws + 2 * N_ATTN;  // [B*H*S]
  float* Lrow = Mrow + (size_t)BB * HH * SS;

  const dim3 blk(256, 1, 1);

  // 1) fused QKV projections (z = 0:Q->ws, 1:K->out, 2:V->out; head-split)
  gemm_proj<true><<<dim3(DD / 64, NROWS / 64, 3), blk, 0, stream>>>(
      hs, wq, wk, wv, bq, bk, bv, Qws, k_out, v_out);

  // 2) raw scores + causal mask + online softmax stats
  attn_scores<<<dim3(SS / 128, BB * HH, 1), blk, 0, stream>>>(Qws, k_out,
                                                              attn_w, Mrow,
                                                              Lrow);

  // 3) normalize weights in place + O = P @ V
  attn_pv<<<dim3(SS / 128, BB * HH, 1), blk, 0, stream>>>(v_out, attn_w, Mrow,
                                                          Lrow, Ows);

  // 4) c_proj
  gemm_proj<false><<<dim3(DD / 64, NROWS / 64, 1), blk, 0, stream>>>(
      Ows, wc, wc, wc, bc, bc, bc, attn_out, attn_out, attn_out);
}